// TopkAttention_70884140253868
// MI455X (gfx1250) — compile-verified
//
#include <hip/hip_runtime.h>
#include <math.h>

#define BH 32
#define NQ 512
#define NKP 8192
#define DIM 128
#define KTOP 64
#define HALF_NK 4096
#define QT 16

typedef __attribute__((ext_vector_type(2))) float v2f;
typedef __attribute__((ext_vector_type(8))) float v8f;

__device__ __forceinline__ unsigned ordkey(float f) {
  unsigned u = __float_as_uint(f);
  return (u & 0x80000000u) ? ~u : (u | 0x80000000u);
}

// ---------------------------------------------------------------------------
// Kernel 1: prefix scores (f32 WMMA) + exact radix top-64 per query.
// Grid: (NQ/16, BH), block 512 (16 waves; one wave per query row for select).
// A-fragments (Q) hoisted to 64 VGPRs; two independent WMMA chains per wave.
// ---------------------------------------------------------------------------
__global__ void __launch_bounds__(512, 1)
topk_scores_kernel(const float* __restrict__ Q,
                   const float* __restrict__ PK,
                   float* __restrict__ tk_exp,
                   int*   __restrict__ tk_idx,
                   float* __restrict__ tk_esum)
{
  const int qtile = blockIdx.x;
  const int bh    = blockIdx.y;
  const int tid   = threadIdx.x;
  const int w     = tid >> 5;
  const int lane  = tid & 31;

  extern __shared__ float smem[];
  float*    sc   = smem;                              // 16*4096 f32
  float*    qt   = sc + QT * HALF_NK;                 // 16*130  f32 (padded)
  unsigned* hist = (unsigned*)(qt + QT * 130);        // 16*256  u32
  float*    cval = (float*)(hist + QT * 256);         // 16*128  f32
  unsigned* cidx = (unsigned*)(cval + QT * 128);      // 16*128  u32
  unsigned* ccnt = cidx + QT * 128;                   // 16*2    u32

  const float* Qg = Q + ((size_t)bh * NQ + (size_t)qtile * QT) * DIM;
  for (int i = tid; i < QT * DIM; i += 512) {
    int m = i >> 7, d = i & 127;
    qt[m * 130 + d] = Qg[m * DIM + d];
  }
  __syncthreads();

  const int    qrow = w;                 // wave's query for selection
  const float* PKb  = PK + (size_t)bh * NKP * DIM;
  const int    mrow = lane & 15;
  const int    khi  = (lane >> 4) << 1;  // K-pair select per half-wave

  // hoist all 32 A-fragments (whole Q row-slice for this lane) into registers
  v2f af[32];
#pragma unroll
  for (int kk = 0; kk < 32; ++kk)
    af[kk] = *(const v2f*)&qt[mrow * 130 + kk * 4 + khi];

  for (int h = 0; h < 2; ++h) {
    const int keybase = h * HALF_NK;

    // ---- 16 x 4096 score tile: two key tiles in flight per wave ----
    for (int t = w; t < HALF_NK / 16; t += 32) {
      const int nA = t * 16;
      const int nB = (t + 16) * 16;
      v8f accA = {}, accB = {};
      const float* krA = PKb + (size_t)(keybase + nA + mrow) * DIM;
      const float* krB = PKb + (size_t)(keybase + nB + mrow) * DIM;
#pragma unroll
      for (int kk = 0; kk < 32; ++kk) {
        v2f bA = *(const v2f*)&krA[kk * 4 + khi];
        v2f bB = *(const v2f*)&krB[kk * 4 + khi];
        accA = __builtin_amdgcn_wmma_f32_16x16x4_f32(
            false, af[kk], false, bA, (short)0, accA, false, false);
        accB = __builtin_amdgcn_wmma_f32_16x16x4_f32(
            false, af[kk], false, bB, (short)0, accB, false, false);
      }
      const int colA = nA + (lane & 15);
      const int colB = nB + (lane & 15);
      const int mb   = (lane >> 4) << 3;
#pragma unroll
      for (int r = 0; r < 8; ++r) {
        sc[(mb + r) * HALF_NK + colA] = accA[r];
        sc[(mb + r) * HALF_NK + colB] = accB[r];
      }
    }
    __syncthreads();

    // ---- exact top-64 of this half via 4-pass radix select ----
    unsigned pfx = 0u, msk = 0u;
    int remaining = KTOP;
    const unsigned hb = (unsigned)w * 256u;
    for (int pass = 0; pass < 4; ++pass) {
      const int shift = 24 - 8 * pass;
      for (int b = lane; b < 256; b += 32) hist[hb + b] = 0u;
      __syncthreads();
      for (int i = 0; i < HALF_NK / 32; ++i) {
        const int j = i * 32 + lane;
        unsigned u = ordkey(sc[qrow * HALF_NK + j]);
        if ((u & msk) == pfx)
          atomicAdd(&hist[hb + ((u >> shift) & 0xFFu)], 1u);
      }
      __syncthreads();
      if (lane == 0) {
        unsigned cum = 0; int bsel = 0;
        for (int b = 255; b >= 0; --b) {
          unsigned hc = hist[hb + b];
          if (cum + hc >= (unsigned)remaining) { bsel = b; break; }
          cum += hc;
        }
        remaining -= (int)cum;
        pfx |= ((unsigned)bsel) << shift;
      }
      pfx = (unsigned)__shfl((int)pfx, 0);
      remaining = __shfl(remaining, 0);
      msk |= 0xFFu << shift;
      __syncthreads();
    }

    const unsigned T  = pfx;
    const int needEq  = remaining;
    const int cGT     = KTOP - needEq;
    if (lane == 0) { ccnt[qrow * 2] = 0u; ccnt[qrow * 2 + 1] = 0u; }
    __syncthreads();

    for (int i = 0; i < HALF_NK / 32; ++i) {
      const int j = i * 32 + lane;
      const float v = sc[qrow * HALF_NK + j];
      const unsigned u = ordkey(v);
      if (u > T) {
        unsigned s = atomicAdd(&ccnt[qrow * 2], 1u);
        cval[qrow * 128 + h * KTOP + s] = v;
        cidx[qrow * 128 + h * KTOP + s] = (unsigned)(keybase + j);
      } else if (u == T) {
        unsigned s = atomicAdd(&ccnt[qrow * 2 + 1], 1u);
        if ((int)s < needEq) {
          cval[qrow * 128 + h * KTOP + cGT + s] = v;
          cidx[qrow * 128 + h * KTOP + cGT + s] = (unsigned)(keybase + j);
        }
      }
    }
    __syncthreads();
  }

  // ---- merge 128 candidates -> global top-64 (64 wave-argmax passes) ----
  float    mv[4];
  unsigned mi[4];
#pragma unroll
  for (int jj = 0; jj < 4; ++jj) {
    mv[jj] = cval[qrow * 128 + jj * 32 + lane];
    mi[jj] = cidx[qrow * 128 + jj * 32 + lane];
  }
  const float  scale = 0.08838834764831845f;  // 1/sqrt(128)
  const size_t qflat = (size_t)bh * NQ + (size_t)qtile * QT + qrow;
  float esum = 0.f;
  const float NEGINF = -__builtin_inff();
  for (int t = 0; t < KTOP; ++t) {
    float vl = fmaxf(fmaxf(mv[0], mv[1]), fmaxf(mv[2], mv[3]));
    float vm = vl;
#pragma unroll
    for (int off = 16; off > 0; off >>= 1) vm = fmaxf(vm, __shfl_xor(vm, off));
    unsigned long long bal = __ballot(vl == vm);
    int src = __ffsll(bal) - 1;
    float e = __expf(vm * scale);
    if (lane == 0) { tk_exp[qflat * KTOP + t] = e; esum += e; }
    if (lane == src) {
      bool done = false;
#pragma unroll
      for (int jj = 0; jj < 4; ++jj) {
        if (!done && mv[jj] == vm) {
          tk_idx[qflat * KTOP + t] = (int)mi[jj];
          mv[jj] = NEGINF;
          done = true;
        }
      }
    }
  }
  if (lane == 0) tk_esum[qflat] = esum;
}

// ---------------------------------------------------------------------------
// Kernel 2: dense causal suffix attention (f32 WMMA), shared denominator,
// sparse gather from prefix V, combine. Grid: (NQ/16, BH), block 256.
// ---------------------------------------------------------------------------
__global__ void __launch_bounds__(256, 1)
dense_combine_kernel(const float* __restrict__ Q,
                     const float* __restrict__ KS,
                     const float* __restrict__ VS,
                     const float* __restrict__ PV,
                     const float* __restrict__ tk_exp,
                     const int*   __restrict__ tk_idx,
                     const float* __restrict__ tk_esum,
                     float* __restrict__ out)
{
  const int qtile = blockIdx.x;
  const int bh    = blockIdx.y;
  const int tid   = threadIdx.x;
  const int w     = tid >> 5;
  const int lane  = tid & 31;

  __shared__ float qt[QT * 130];
  __shared__ float sc[QT * 516];
  __shared__ float outb[QT * 132];
  __shared__ float invden[QT];

  const float  scale = 0.08838834764831845f;
  const int    q0    = qtile * QT;
  const float* Qg  = Q  + ((size_t)bh * NQ + q0) * DIM;
  const float* KSb = KS + (size_t)bh * NQ * DIM;
  const float* VSb = VS + (size_t)bh * NQ * DIM;

  for (int i = tid; i < QT * DIM; i += 256) {
    int m = i >> 7, d = i & 127;
    qt[m * 130 + d] = Qg[m * DIM + d];
  }
  __syncthreads();

  const int mrow = lane & 15;
  const int khi  = (lane >> 4) << 1;

  // hoisted Q fragments
  v2f af[32];
#pragma unroll
  for (int kk = 0; kk < 32; ++kk)
    af[kk] = *(const v2f*)&qt[mrow * 130 + kk * 4 + khi];

  // ---- masked exp(score) tile 16x512: two key tiles in flight ----
  for (int t = w; t < NQ / 16; t += 16) {
    const int nA = t * 16;
    const int nB = (t + 8) * 16;
    v8f accA = {}, accB = {};
    const float* krA = KSb + (size_t)(nA + mrow) * DIM;
    const float* krB = KSb + (size_t)(nB + mrow) * DIM;
#pragma unroll
    for (int kk = 0; kk < 32; ++kk) {
      v2f bA = *(const v2f*)&krA[kk * 4 + khi];
      v2f bB = *(const v2f*)&krB[kk * 4 + khi];
      accA = __builtin_amdgcn_wmma_f32_16x16x4_f32(
          false, af[kk], false, bA, (short)0, accA, false, false);
      accB = __builtin_amdgcn_wmma_f32_16x16x4_f32(
          false, af[kk], false, bB, (short)0, accB, false, false);
    }
    const int colA = nA + (lane & 15);
    const int colB = nB + (lane & 15);
    const int mb   = (lane >> 4) << 3;
#pragma unroll
    for (int r = 0; r < 8; ++r) {
      const int m = mb + r;
      sc[m * 516 + colA] = (colA <= q0 + m) ? __expf(accA[r] * scale) : 0.f;
      sc[m * 516 + colB] = (colB <= q0 + m) ? __expf(accB[r] * scale) : 0.f;
    }
  }
  __syncthreads();

  // ---- row sums + shared denominator ----
  for (int rr = w; rr < QT; rr += 8) {
    float s = 0.f;
    for (int i = 0; i < NQ / 32; ++i) s += sc[rr * 516 + i * 32 + lane];
#pragma unroll
    for (int off = 16; off > 0; off >>= 1) s += __shfl_xor(s, off);
    if (lane == 0) {
      const size_t qflat = (size_t)bh * NQ + q0 + rr;
      invden[rr] = 1.f / (s + tk_esum[qflat]);
    }
  }
  __syncthreads();

  // ---- dense numerator @ V_suffix : wave w -> output cols [16w, 16w+16) ----
  {
    const int n0 = w * 16;
    v8f acc0 = {}, acc1 = {};   // two chains over split-K
    for (int k0 = 0; k0 < NQ; k0 += 8) {
      v2f a0 = *(const v2f*)&sc[mrow * 516 + k0 + khi];
      v2f a1 = *(const v2f*)&sc[mrow * 516 + k0 + 4 + khi];
      v2f b0, b1;
      b0.x = VSb[(size_t)(k0 + khi) * DIM + n0 + mrow];
      b0.y = VSb[(size_t)(k0 + khi + 1) * DIM + n0 + mrow];
      b1.x = VSb[(size_t)(k0 + 4 + khi) * DIM + n0 + mrow];
      b1.y = VSb[(size_t)(k0 + 4 + khi + 1) * DIM + n0 + mrow];
      acc0 = __builtin_amdgcn_wmma_f32_16x16x4_f32(
          false, a0, false, b0, (short)0, acc0, false, false);
      acc1 = __builtin_amdgcn_wmma_f32_16x16x4_f32(
          false, a1, false, b1, (short)0, acc1, false, false);
    }
    const int mb = (lane >> 4) << 3;
#pragma unroll
    for (int r = 0; r < 8; ++r)
      outb[(mb + r) * 132 + n0 + (lane & 15)] = acc0[r] + acc1[r];
  }
  __syncthreads();

  // ---- sparse gather from prefix V + combine + normalize ----
  const float* PVb = PV + (size_t)bh * NKP * DIM;
  for (int rr = w; rr < QT; rr += 8) {
    const size_t qflat = (size_t)bh * NQ + q0 + rr;
    const int d0 = lane * 4;
    float o0 = outb[rr * 132 + d0 + 0];
    float o1 = outb[rr * 132 + d0 + 1];
    float o2 = outb[rr * 132 + d0 + 2];
    float o3 = outb[rr * 132 + d0 + 3];
    for (int k = 0; k < KTOP; ++k) {
      const float wgt = tk_exp[qflat * KTOP + k];
      const int   idx = tk_idx[qflat * KTOP + k];
      const float4 pv = *(const float4*)&PVb[(size_t)idx * DIM + d0];
      o0 += wgt * pv.x; o1 += wgt * pv.y; o2 += wgt * pv.z; o3 += wgt * pv.w;
    }
    const float inv = invden[rr];
    float4 r4 = make_float4(o0 * inv, o1 * inv, o2 * inv, o3 * inv);
    *(float4*)&out[qflat * DIM + d0] = r4;
  }
}

// ---------------------------------------------------------------------------
extern "C" void kernel_launch(void* const* d_in, const int* in_sizes, int n_in,
                              void* d_out, int out_size, void* d_ws, size_t ws_size,
                              hipStream_t stream) {
  (void)in_sizes; (void)n_in; (void)out_size; (void)ws_size;
  const float* Q  = (const float*)d_in[0];
  const float* KS = (const float*)d_in[1];
  const float* VS = (const float*)d_in[2];
  const float* PK = (const float*)d_in[3];
  const float* PV = (const float*)d_in[4];
  float* out = (float*)d_out;

  // workspace: exp-values (4MB) | indices (4MB) | exp-sums (64KB)
  float* tk_exp  = (float*)d_ws;
  int*   tk_idx  = (int*)((char*)d_ws + (size_t)BH * NQ * KTOP * 4);
  float* tk_esum = (float*)((char*)d_ws + 2 * (size_t)BH * NQ * KTOP * 4);

  const size_t smem1 =
      (size_t)(QT * HALF_NK + QT * 130 + QT * 256 + QT * 128 * 2 + QT * 2) * 4;
  (void)hipFuncSetAttribute((const void*)topk_scores_kernel,
                            hipFuncAttributeMaxDynamicSharedMemorySize,
                            (int)smem1);

  topk_scores_kernel<<<dim3(NQ / QT, BH), 512, smem1, stream>>>(
      Q, PK, tk_exp, tk_idx, tk_esum);
  dense_combine_kernel<<<dim3(NQ / QT, BH), 256, 0, stream>>>(
      Q, KS, VS, PV, tk_exp, tk_idx, tk_esum, out);
}